// HyperMLPPrediction_54013508715066
// MI455X (gfx1250) — compile-verified
//
#include <hip/hip_runtime.h>
#include <hip/hip_bf16.h>

// ---------------------------------------------------------------------------
// HyperMLP prediction for MI455X (gfx1250, wave32).
// Memory-bound (0.5 FLOP/byte): stream per-sample weights once (NT loads),
// compute batched GEMV via V_WMMA_F32_16X16X4_F32, activations in LDS.
// ---------------------------------------------------------------------------

typedef __attribute__((ext_vector_type(2))) float v2f;
typedef __attribute__((ext_vector_type(8))) float v8f;

#define HMLP_TOTAL_PARAMS 657024
#define HMLP_BATCH        512

// Per-sample parameter offsets inside a z row:
//   L0: w@0      (128x512), b@65536  (512)
//   L1: w@66048  (512x512), b@328192 (512)
//   L2: w@328704 (512x512), b@590848 (512)
//   L3: w@591360 (512x128), b@656896 (128)

template <int DIN, int DOUT, bool RELU>
__device__ __forceinline__ void hmlp_layer(
    const float* __restrict__ w,     // (DIN, DOUT) row-major, this sample
    const float* __restrict__ bias,  // (DOUT)
    const float* __restrict__ xin,   // LDS: DIN floats
    float* __restrict__ xout,        // LDS (or global for last layer): DOUT floats
    int wave, int lane)
{
    constexpr int NT  = DOUT / 16;   // 16-wide output tiles
    constexpr int TPW = NT / 8;      // tiles per wave (8 waves per block)
    constexpr int NK  = DIN / 4;     // K-chunks of 4

    const int m    = lane & 15;             // output column within tile (A rows)
    const int mlo  = (lane < 16) ? 0 : 8;   // C/D: reg r holds M = r + mlo
    const int koff = (lane < 16) ? 0 : 2;   // A/B: reg j holds K = j + koff

    // Accumulators start as bias (C = bias broadcast across N).
    v8f acc[TPW];
#pragma unroll
    for (int t = 0; t < TPW; ++t) {
        const int o0 = (wave + 8 * t) * 16;
#pragma unroll
        for (int r = 0; r < 8; ++r)
            acc[t][r] = bias[o0 + mlo + r];
    }

    for (int kc = 0; kc < NK; ++kc) {
        const int i0 = kc * 4;

        // B fragment: x[i0..i0+3] broadcast across all 16 N columns.
        v2f bf;
        bf.x = xin[i0 + koff];
        bf.y = xin[i0 + koff + 1];

        // Stream-ahead prefetch of weight rows (~8 K-chunks ahead).
        if (kc + 8 < NK)
            __builtin_prefetch(w + (size_t)(i0 + 32) * DOUT + wave * 16, 0, 0);

#pragma unroll
        for (int t = 0; t < TPW; ++t) {
            const int o0 = (wave + 8 * t) * 16;
            // A fragment: w^T tile — A[m][k] = w[i0+k][o0+m].
            // Weights are streamed once: use non-temporal loads.
            v2f af;
            af.x = __builtin_nontemporal_load(
                       w + (size_t)(i0 + koff) * DOUT + o0 + m);
            af.y = __builtin_nontemporal_load(
                       w + (size_t)(i0 + koff + 1) * DOUT + o0 + m);

            // D = A x B + C  (8 args: neg_a, A, neg_b, B, c_mod, C, reuse_a, reuse_b)
            acc[t] = __builtin_amdgcn_wmma_f32_16x16x4_f32(
                         false, af, false, bf, (short)0, acc[t], false, false);
        }
    }

    // All N columns of D are identical (B was broadcast). D reg r holds
    // M = r (lanes 0-15) / r+8 (lanes 16-31); lanes 0 and 16 write results.
    if ((lane & 15) == 0) {
#pragma unroll
        for (int t = 0; t < TPW; ++t) {
            const int o0 = (wave + 8 * t) * 16;
#pragma unroll
            for (int r = 0; r < 8; ++r) {
                float v = acc[t][r];
                if (RELU) v = fmaxf(v, 0.0f);
                xout[o0 + mlo + r] = v;
            }
        }
    }
}

__global__ __launch_bounds__(256) void HyperMLPPrediction_kernel(
    const float* __restrict__ z,    // (512, 657024)
    const float* __restrict__ xq,   // (512, 128)
    float* __restrict__ out)        // (512, 128)
{
    __shared__ float xb[2][512];    // double-buffered activations (4 KB)

    const int b   = blockIdx.x;     // one workgroup per sample: contiguous 2.6 MB row
    const int tid = threadIdx.x;
    const float* zb = z + (size_t)b * HMLP_TOTAL_PARAMS;

    if (tid < 128) xb[0][tid] = xq[b * 128 + tid];
    __syncthreads();

    const int wave = tid >> 5;
    const int lane = tid & 31;

    hmlp_layer<128, 512, true >(zb + 0,      zb + 65536,  xb[0], xb[1], wave, lane);
    __syncthreads();
    hmlp_layer<512, 512, true >(zb + 66048,  zb + 328192, xb[1], xb[0], wave, lane);
    __syncthreads();
    hmlp_layer<512, 512, true >(zb + 328704, zb + 590848, xb[0], xb[1], wave, lane);
    __syncthreads();
    hmlp_layer<512, 128, false>(zb + 591360, zb + 656896, xb[1],
                                out + (size_t)b * 128, wave, lane);
}

extern "C" void kernel_launch(void* const* d_in, const int* in_sizes, int n_in,
                              void* d_out, int out_size, void* d_ws, size_t ws_size,
                              hipStream_t stream) {
    const float* z  = (const float*)d_in[0];   // (512, 657024) f32
    const float* xq = (const float*)d_in[1];   // (512, 128)    f32
    float* out = (float*)d_out;                // (512, 128)    f32
    (void)in_sizes; (void)n_in; (void)out_size; (void)d_ws; (void)ws_size;

    HyperMLPPrediction_kernel<<<HMLP_BATCH, 256, 0, stream>>>(z, xq, out);
}